// ConvexPlaneComputer_52132313039480
// MI455X (gfx1250) — compile-verified
//
#include <hip/hip_runtime.h>

typedef float v2f __attribute__((ext_vector_type(2)));
typedef float v8f __attribute__((ext_vector_type(8)));

#define N_PTS    200000
#define TILES    12500          // N_PTS / 16
#define M_TGT    256
#define N_GROUPS 16             // 16 targets per group
#define ITERS    50
#define SLICES   64             // blocks per target group
#define THREADS  256            // 8 waves per block
#define WPB      8              // waves per block
#define WPG      (SLICES * WPB) // 512 waves per target group
#define EPS_SEP  (-1e-6f)
#define EPS_NORM (1e-8f)

__device__ __forceinline__ unsigned umin32(unsigned a, unsigned b) { return a < b ? a : b; }

// Pass over all point tiles: WMMA separation-dot (mask update) + WMMA distance
// (with t^2 folded into the accumulator) + packed-key integer argmin.
// One lane owns one target column and 8 point rows per tile
// (C/D layout of V_WMMA_F32_16X16X4_F32).
__global__ __launch_bounds__(THREADS)
void planes_scan_kernel(const float* __restrict__ pc,
                        const float* __restrict__ tg,
                        const float* __restrict__ planes,   // [256][4] ax,ay,az,b
                        unsigned char* __restrict__ mask,   // [16][TILES][32]
                        unsigned* __restrict__ part_d,      // [16][WPG][16] packed keys
                        int*      __restrict__ part_i,
                        int iter)
{
    const int g    = blockIdx.x >> 6;        // target group (SLICES==64)
    const int s    = blockIdx.x & (SLICES - 1);
    const int lane = threadIdx.x & 31;
    const int wave = threadIdx.x >> 5;
    const int gw   = s * WPB + wave;         // wave id within group
    const int n    = lane & 15;              // target column owned by this lane
    const int hi   = lane >> 4;              // 0: K=0,1 rows 0..7 ; 1: K=2,3 rows 8..15
    const int tgt  = g * 16 + n;

    // Stationary B matrices (4x16): lanes 0-15 hold K=0,1; lanes 16-31 hold K=2,3.
    const float tx = tg[tgt * 3 + 0];
    const float ty = tg[tgt * 3 + 1];
    const float tz = tg[tgt * 3 + 2];
    const float t2 = tx * tx + ty * ty + tz * tz;
    v2f Bd, Bp;
    if (hi == 0) { Bd.x = -2.0f * tx; Bd.y = -2.0f * ty; }
    else         { Bd.x = -2.0f * tz; Bd.y = 1.0f; }
    Bp.x = 0.0f; Bp.y = 0.0f;
    if (iter > 0) {
        if (hi == 0) { Bp.x = planes[tgt * 4 + 0]; Bp.y =  planes[tgt * 4 + 1]; }
        else         { Bp.x = planes[tgt * 4 + 2]; Bp.y = -planes[tgt * 4 + 3]; }
    }
    // C accumulator carrying t^2 -> WMMA emits true (non-negative) squared distance.
    v8f Ct2;
#pragma unroll
    for (int r = 0; r < 8; ++r) Ct2[r] = t2;

    unsigned bestdk = 0xFFFFFFFFu;           // packed distance key (bits of d2, low 3 cleared)
    int      bestidx = 0;
    const int rowbase = hi << 3;

    for (int t = gw; t < TILES; t += WPG) {
        const float* p = pc + (size_t)(t * 16 + n) * 3;
        const float x = p[0], y = p[1], z = p[2];
        // A matrices (16x4): lanes 0-15 hold K=0,1; lanes 16-31 hold K=2,3.
        v2f Ad, Ap;
        if (hi == 0) { Ad.x = x; Ad.y = y;                     Ap.x = x; Ap.y = y;    }
        else         { Ad.x = z; Ad.y = x * x + y * y + z * z; Ap.x = z; Ap.y = 1.0f; }

        unsigned char* mp = mask + (((size_t)(g * TILES + t)) << 5) + lane;
        unsigned mk;
        if (iter == 0) {
            mk = 0xFFu;                       // all alive, nothing stored yet
        } else {
            mk = (iter == 1) ? 0xFFu : (unsigned)(*mp);
            // separation: dot[m][n] = a.p - b  (previous iteration's plane)
            v8f c0 = {};
            v8f dotv = __builtin_amdgcn_wmma_f32_16x16x4_f32(
                false, Ap, false, Bp, (short)0, c0, false, false);
#pragma unroll
            for (int r = 0; r < 8; ++r)
                if (dotv[r] >= EPS_SEP) mk &= ~(1u << r);
            *mp = (unsigned char)mk;
        }

        // distance: d2[m][n] = t2 + |p|^2 - 2 t.p  (>= 0 -> bit pattern is order-preserving)
        v8f d2 = __builtin_amdgcn_wmma_f32_16x16x4_f32(
            false, Ad, false, Bd, (short)0, Ct2, false, false);

        const unsigned dead = ~mk;
        unsigned k0, k1, k2, k3, k4, k5, k6, k7;
        {
            // key = (bits(d2) & ~7) | r, dead entries get the sign bit (unsigned-max).
            k0 = ((__float_as_uint(d2[0]) & ~7u) | 0u) | (((dead >> 0) & 1u) << 31);
            k1 = ((__float_as_uint(d2[1]) & ~7u) | 1u) | (((dead >> 1) & 1u) << 31);
            k2 = ((__float_as_uint(d2[2]) & ~7u) | 2u) | (((dead >> 2) & 1u) << 31);
            k3 = ((__float_as_uint(d2[3]) & ~7u) | 3u) | (((dead >> 3) & 1u) << 31);
            k4 = ((__float_as_uint(d2[4]) & ~7u) | 4u) | (((dead >> 4) & 1u) << 31);
            k5 = ((__float_as_uint(d2[5]) & ~7u) | 5u) | (((dead >> 5) & 1u) << 31);
            k6 = ((__float_as_uint(d2[6]) & ~7u) | 6u) | (((dead >> 6) & 1u) << 31);
            k7 = ((__float_as_uint(d2[7]) & ~7u) | 7u) | (((dead >> 7) & 1u) << 31);
        }
        const unsigned tkey = umin32(umin32(umin32(k0, k1), umin32(k2, k3)),
                                     umin32(umin32(k4, k5), umin32(k6, k7)));
        const unsigned dk  = tkey & ~7u;                 // strip r for cross-tile tie-break
        const int      idx = t * 16 + rowbase + (int)(tkey & 7u);
        if (dk < bestdk) { bestdk = dk; bestidx = idx; } // strict < => first (earlier tile) min
    }

    // Combine the two half-wave rows of the same target column.
    const unsigned od = (unsigned)__shfl_xor((int)bestdk, 16, 32);
    const int      oi = __shfl_xor(bestidx, 16, 32);
    if (od < bestdk || (od == bestdk && oi < bestidx)) { bestdk = od; bestidx = oi; }

    if (hi == 0) {
        part_d[(g * WPG + gw) * 16 + n] = bestdk;
        part_i[(g * WPG + gw) * 16 + n] = bestidx;
    }
}

// Fold wave partials per target, compute the plane, emit outputs.
__global__ __launch_bounds__(THREADS)
void planes_reduce_kernel(const float* __restrict__ pc,
                          const float* __restrict__ tg,
                          const unsigned* __restrict__ part_d,
                          const int*      __restrict__ part_i,
                          float* __restrict__ planes,
                          float* __restrict__ out_a,   // [50][256][3]
                          float* __restrict__ out_b,   // [50][256]
                          int iter)
{
    const int tgt = threadIdx.x;
    if (tgt >= M_TGT) return;
    const int g = tgt >> 4, n = tgt & 15;

    unsigned best = 0xFFFFFFFFu;
    int      bidx = 0;
    const unsigned* pd = part_d + (size_t)g * WPG * 16 + n;
    const int*      pi = part_i + (size_t)g * WPG * 16 + n;
    for (int w = 0; w < WPG; ++w) {
        const unsigned d = pd[w * 16];
        const int      i = pi[w * 16];
        if (d < best || (d == best && i < bidx)) { best = d; bidx = i; }
    }
    // All entries masked (sign bit set) -> argmin over all-inf row is 0 (jnp semantics).
    if (best & 0x80000000u) bidx = 0;

    const float cx = pc[bidx * 3 + 0];
    const float cy = pc[bidx * 3 + 1];
    const float cz = pc[bidx * 3 + 2];
    float ax = cx - tg[tgt * 3 + 0];
    float ay = cy - tg[tgt * 3 + 1];
    float az = cz - tg[tgt * 3 + 2];
    const float nrm = __builtin_sqrtf(ax * ax + ay * ay + az * az) + EPS_NORM;
    ax /= nrm; ay /= nrm; az /= nrm;
    const float b = ax * cx + ay * cy + az * cz;

    planes[tgt * 4 + 0] = ax;
    planes[tgt * 4 + 1] = ay;
    planes[tgt * 4 + 2] = az;
    planes[tgt * 4 + 3] = b;

    out_a[((size_t)iter * M_TGT + tgt) * 3 + 0] = ax;
    out_a[((size_t)iter * M_TGT + tgt) * 3 + 1] = ay;
    out_a[((size_t)iter * M_TGT + tgt) * 3 + 2] = az;
    out_b[(size_t)iter * M_TGT + tgt] = b;
}

extern "C" void kernel_launch(void* const* d_in, const int* in_sizes, int n_in,
                              void* d_out, int out_size, void* d_ws, size_t ws_size,
                              hipStream_t stream) {
    (void)in_sizes; (void)n_in; (void)out_size; (void)ws_size;
    const float* pc = (const float*)d_in[0];   // [200000,3]
    const float* tg = (const float*)d_in[1];   // [256,3]

    float* out_a = (float*)d_out;                       // 50*256*3 floats
    float* out_b = out_a + (size_t)ITERS * M_TGT * 3;   // 50*256 floats

    // workspace carve-out (~7.5 MB total, all L2-resident)
    unsigned char* ws   = (unsigned char*)d_ws;
    unsigned char* mask = ws;                                        // 16*12500*32 = 6.4 MB
    size_t off = ((size_t)N_GROUPS * TILES * 32 + 255) & ~(size_t)255;
    unsigned* part_d = (unsigned*)(ws + off); off += (size_t)N_GROUPS * WPG * 16 * sizeof(unsigned);
    int*      part_i = (int*)     (ws + off); off += (size_t)N_GROUPS * WPG * 16 * sizeof(int);
    float*    planes = (float*)   (ws + off);                        // 256*4 floats

    for (int iter = 0; iter < ITERS; ++iter) {
        planes_scan_kernel<<<dim3(N_GROUPS * SLICES), dim3(THREADS), 0, stream>>>(
            pc, tg, planes, mask, part_d, part_i, iter);
        planes_reduce_kernel<<<dim3(1), dim3(THREADS), 0, stream>>>(
            pc, tg, part_d, part_i, planes, out_a, out_b, iter);
    }
}